// GlobalPool3d_54640573939778
// MI455X (gfx1250) — compile-verified
//
#include <hip/hip_runtime.h>

typedef __attribute__((ext_vector_type(2))) float v2f;
typedef __attribute__((ext_vector_type(8))) float v8f;

#define CH    128
#define CHUNK 4096   // rows per phase-1 block

// ---------------- Phase 0: prefix-sum of segment lengths (B=16, trivial) ----
__global__ void seg_offsets_kernel(const int* __restrict__ nv, int B,
                                   int* __restrict__ offs) {
    if (blockIdx.x == 0 && threadIdx.x == 0) {
        int acc = 0;
        offs[0] = 0;
        for (int b = 0; b < B; ++b) { acc += nv[b]; offs[b + 1] = acc; }
    }
}

// ---------------- Phase 1: per-(segment, chunk) partial column sums ---------
// Block = 256 threads = 8 waves; wave w owns channels [16w, 16w+16).
// Column sum via  D = Ones(16x4) * B(4x16) + C  with V_WMMA_F32_16X16X4_F32:
// every row of D is the 4-row column sum, so C chains as the fp32 accumulator.
__global__ void __launch_bounds__(256)
seg_partial_wmma_kernel(const float* __restrict__ in, const int* __restrict__ offs,
                        float* __restrict__ partials, int chunks) {
    const int j   = blockIdx.x;   // chunk index within segment
    const int b   = blockIdx.y;   // segment
    const int tid = threadIdx.x;

    float* out = partials + ((size_t)b * chunks + j) * CH;

    const int segStart = offs[b];
    const int segEnd   = offs[b + 1];
    const int start    = segStart + j * CHUNK;
    int end = start + CHUNK;
    if (end > segEnd) end = segEnd;

    if (start >= segEnd) {            // chunk beyond this segment: zero slot
        if (tid < CH) out[tid] = 0.0f;
        return;
    }

    const int wave = tid >> 5;        // 0..7 -> channel group
    const int lane = tid & 31;
    const int half = lane >> 4;       // 0 or 1 (K-pair select)
    const int n    = lane & 15;       // column within the 16-wide tile
    const int c    = (wave << 4) + n; // this lane's channel

    const v2f ones = {1.0f, 1.0f};
    v8f acc0 = {0.f, 0.f, 0.f, 0.f, 0.f, 0.f, 0.f, 0.f};
    v8f acc1 = acc0, acc2 = acc0, acc3 = acc0;

    // Per-lane walking pointer: row (start + 2*half), channel c.
    const float* p = in + (size_t)start * CH + (size_t)(half << 1) * CH + c;

    // ---- main loop: 16 rows/iter, no bounds checks, 8 loads in flight ----
    const int nfull = (end - start) & ~15;
    const int full  = start + nfull;
    for (int r = start; r < full; r += 16) {
        float x0 = __builtin_nontemporal_load(p + 0 * CH);
        float x1 = __builtin_nontemporal_load(p + 1 * CH);
        float x2 = __builtin_nontemporal_load(p + 4 * CH);
        float x3 = __builtin_nontemporal_load(p + 5 * CH);
        float x4 = __builtin_nontemporal_load(p + 8 * CH);
        float x5 = __builtin_nontemporal_load(p + 9 * CH);
        float x6 = __builtin_nontemporal_load(p + 12 * CH);
        float x7 = __builtin_nontemporal_load(p + 13 * CH);
        p += 16 * CH;
        v2f b0 = {x0, x1};
        v2f b1 = {x2, x3};
        v2f b2 = {x4, x5};
        v2f b3 = {x6, x7};
        acc0 = __builtin_amdgcn_wmma_f32_16x16x4_f32(false, ones, false, b0,
                                                     (short)0, acc0, false, false);
        acc1 = __builtin_amdgcn_wmma_f32_16x16x4_f32(false, ones, false, b1,
                                                     (short)0, acc1, false, false);
        acc2 = __builtin_amdgcn_wmma_f32_16x16x4_f32(false, ones, false, b2,
                                                     (short)0, acc2, false, false);
        acc3 = __builtin_amdgcn_wmma_f32_16x16x4_f32(false, ones, false, b3,
                                                     (short)0, acc3, false, false);
    }

    // ---- tail: up to 15 rows, guarded per-lane loads (zeros are inert) ----
    for (int r = full; r < end; r += 8) {
        const int ra = r + (half << 1);
        const int rb = ra + 4;
        float x0 = 0.f, x1 = 0.f, x2 = 0.f, x3 = 0.f;
        if (ra     < end) x0 = __builtin_nontemporal_load(&in[(size_t)(ra    ) * CH + c]);
        if (ra + 1 < end) x1 = __builtin_nontemporal_load(&in[(size_t)(ra + 1) * CH + c]);
        if (rb     < end) x2 = __builtin_nontemporal_load(&in[(size_t)(rb    ) * CH + c]);
        if (rb + 1 < end) x3 = __builtin_nontemporal_load(&in[(size_t)(rb + 1) * CH + c]);
        v2f bA = {x0, x1};
        v2f bB = {x2, x3};
        acc0 = __builtin_amdgcn_wmma_f32_16x16x4_f32(false, ones, false, bA,
                                                     (short)0, acc0, false, false);
        acc1 = __builtin_amdgcn_wmma_f32_16x16x4_f32(false, ones, false, bB,
                                                     (short)0, acc1, false, false);
    }

    v8f acc = (acc0 + acc1) + (acc2 + acc3);

    // D layout: VGPR0, lanes 0..15 hold M=0, N=lane -> column sums.
    if (lane < 16) out[c] = acc[0];
}

// ---------------- Phase 2: deterministic reduction of partials + mean -------
__global__ void seg_finalize_kernel(const float* __restrict__ partials,
                                    const int* __restrict__ nv,
                                    float* __restrict__ out, int chunks, int B) {
    const int t = blockIdx.x * blockDim.x + threadIdx.x;
    if (t >= B * CH) return;
    const int b = t / CH;
    const int c = t % CH;
    const float* p = partials + (size_t)b * chunks * CH + c;
    float s = 0.0f;
    for (int jj = 0; jj < chunks; ++jj) s += p[(size_t)jj * CH];
    float cnt = (float)nv[b];
    if (cnt < 1.0f) cnt = 1.0f;
    out[t] = s / cnt;
}

// ---------------- Launch ----------------------------------------------------
extern "C" void kernel_launch(void* const* d_in, const int* in_sizes, int n_in,
                              void* d_out, int out_size, void* d_ws, size_t ws_size,
                              hipStream_t stream) {
    const float* in  = (const float*)d_in[0];
    const int*   nv  = (const int*)d_in[1];
    float*       out = (float*)d_out;

    const int B      = in_sizes[1];
    const int N      = in_sizes[0] / CH;
    const int chunks = (N + CHUNK - 1) / CHUNK;

    int*   offs     = (int*)d_ws;                        // (B+1) ints
    float* partials = (float*)((char*)d_ws + 512);       // B*chunks*CH floats

    seg_offsets_kernel<<<1, 32, 0, stream>>>(nv, B, offs);

    dim3 grid(chunks, B);
    seg_partial_wmma_kernel<<<grid, 256, 0, stream>>>(in, offs, partials, chunks);

    const int tot = B * CH;
    seg_finalize_kernel<<<(tot + 255) / 256, 256, 0, stream>>>(partials, nv, out,
                                                              chunks, B);
}